// main_model_43525198578059
// MI455X (gfx1250) — compile-verified
//
#include <hip/hip_runtime.h>
#include <hip/hip_bf16.h>

// ---------------------------------------------------------------------------
// Hi-LANDER style GNN: 4x GAT layers + edge MLP head, for MI455X (gfx1250).
// GEMMs use V_WMMA_F32_16X16X4_F32 (fp32 matrix pipe, wave32).
// Segment ops use a CSR-by-dst built once per call (no fp32 atomics).
// ---------------------------------------------------------------------------

typedef float v2f __attribute__((ext_vector_type(2)));
typedef float v8f __attribute__((ext_vector_type(8)));

constexpr int cN = 50000;          // nodes
constexpr int cE = 800000;         // edges
constexpr int cF = 256;            // feature dim
constexpr int cNH = 128;           // head hidden dim
constexpr float NEG_SLOPE = 0.2f;

// ---------------------------------------------------------------- utilities
__global__ void k_zero_i32(int* p, int n) {
    int i = blockIdx.x * blockDim.x + threadIdx.x;
    if (i < n) p[i] = 0;
}

__global__ void k_concat(const float* __restrict__ f, const float* __restrict__ cf,
                         float* __restrict__ x0) {
    int i = blockIdx.x * blockDim.x + threadIdx.x;
    if (i >= cN * 2 * cF) return;
    int n = i / (2 * cF);
    int c = i % (2 * cF);
    x0[i] = (c < cF) ? f[n * cF + c] : cf[n * cF + (c - cF)];
}

// ------------------------------------------------------------- CSR building
__global__ void k_deg(const int* __restrict__ dst, int* __restrict__ deg) {
    int e = blockIdx.x * blockDim.x + threadIdx.x;
    if (e < cE) atomicAdd(&deg[dst[e]], 1);
}

// single-workgroup exclusive scan (1024 threads, serial over chunks)
__global__ void k_scan(const int* __restrict__ deg, int* __restrict__ off, int n) {
    __shared__ int sdata[1024];
    __shared__ int sbase;
    if (threadIdx.x == 0) sbase = 0;
    __syncthreads();
    int nch = (n + 1023) / 1024;
    for (int c = 0; c < nch; ++c) {
        int i = c * 1024 + threadIdx.x;
        int v = (i < n) ? deg[i] : 0;
        sdata[threadIdx.x] = v;
        __syncthreads();
        for (int ofs = 1; ofs < 1024; ofs <<= 1) {
            int t = (threadIdx.x >= (unsigned)ofs) ? sdata[threadIdx.x - ofs] : 0;
            __syncthreads();
            sdata[threadIdx.x] += t;
            __syncthreads();
        }
        if (i < n) off[i] = sbase + sdata[threadIdx.x] - v;   // exclusive
        int chunk_total = sdata[1023];
        __syncthreads();
        if (threadIdx.x == 0) sbase += chunk_total;
        __syncthreads();
    }
    if (threadIdx.x == 0) off[n] = sbase;
}

__global__ void k_place(const int* __restrict__ dst, const int* __restrict__ off,
                        int* __restrict__ cursor, int* __restrict__ csr) {
    int e = blockIdx.x * blockDim.x + threadIdx.x;
    if (e >= cE) return;
    int d = dst[e];
    int pos = atomicAdd(&cursor[d], 1);
    csr[off[d] + pos] = e;
}

// -------------------------------------------------- WMMA fp32 GEMM (generic)
// C[cN x M] = A[cN x K] @ W[K x M] (+ bias). One wave per 16x16 output tile.
__global__ void k_gemm(const float* __restrict__ A, const float* __restrict__ W,
                       const float* __restrict__ bias, float* __restrict__ C,
                       int K, int M) {
    int wave  = (blockIdx.x * blockDim.x + threadIdx.x) >> 5;
    int ntile = M >> 4;
    int tiles = (cN / 16) * ntile;
    if (wave >= tiles) return;                 // wave-uniform
    int lane = threadIdx.x & 31;
    int half = lane >> 4;
    int l16  = lane & 15;
    int tm = wave / ntile;
    int tn = wave % ntile;
    const float* Ap = A + (size_t)(tm * 16 + l16) * K;
    int col = tn * 16 + l16;

    v8f acc = {0.f, 0.f, 0.f, 0.f, 0.f, 0.f, 0.f, 0.f};
    for (int k0 = 0; k0 < K; k0 += 4) {
        int ka = k0 + half * 2;
        v2f a, b;
        a[0] = Ap[ka];
        a[1] = Ap[ka + 1];
        b[0] = W[(size_t)ka * M + col];
        b[1] = W[(size_t)(ka + 1) * M + col];
        acc = __builtin_amdgcn_wmma_f32_16x16x4_f32(
            false, a, false, b, (short)0, acc, false, false);
    }
    float bv = bias ? bias[col] : 0.f;
#pragma unroll
    for (int r = 0; r < 8; ++r) {
        int orow = tm * 16 + half * 8 + r;
        C[(size_t)orow * M + col] = acc[r] + bv;
    }
}

// -------------------------------------------------------- GAT layer pieces
__global__ void k_scores(const float* __restrict__ h, const float* __restrict__ al,
                         const float* __restrict__ ar, float* __restrict__ el,
                         float* __restrict__ er, int M) {
    int n = blockIdx.x * blockDim.x + threadIdx.x;
    if (n >= cN) return;
    const float* hp = h + (size_t)n * M;
    float sl = 0.f, sr = 0.f;
    for (int c = 0; c < M; ++c) {
        float v = hp[c];
        sl += v * al[c];
        sr += v * ar[c];
    }
    el[n] = sl;
    er[n] = sr;
}

__global__ void k_logits(const float* __restrict__ el, const float* __restrict__ er,
                         const int* __restrict__ src, const int* __restrict__ dst,
                         float* __restrict__ e_edge) {
    int e = blockIdx.x * blockDim.x + threadIdx.x;
    if (e >= cE) return;
    float v = el[src[e]] + er[dst[e]];
    e_edge[e] = (v >= 0.f) ? v : NEG_SLOPE * v;
}

__global__ void k_segmax_z(const float* __restrict__ e_edge, const int* __restrict__ csr,
                           const int* __restrict__ off, float* __restrict__ mb,
                           float* __restrict__ zb) {
    int n = blockIdx.x * blockDim.x + threadIdx.x;
    if (n >= cN) return;
    int s0 = off[n], s1 = off[n + 1];
    float m = -1e30f;
    for (int i = s0; i < s1; ++i) m = fmaxf(m, e_edge[csr[i]]);
    float z = 0.f;
    for (int i = s0; i < s1; ++i) z += __expf(e_edge[csr[i]] - m);
    mb[n] = m;
    zb[n] = z;
}

__global__ void k_alpha(const float* __restrict__ e_edge, const float* __restrict__ mb,
                        const float* __restrict__ zb, const int* __restrict__ dst,
                        float* __restrict__ alpha) {
    int e = blockIdx.x * blockDim.x + threadIdx.x;
    if (e >= cE) return;
    int d = dst[e];
    alpha[e] = __expf(e_edge[e] - mb[d]) / zb[d];
}

// out[n] = relu( sum_{e in bucket(n)} alpha[e] * h[src[e]] + b ).  Wave/node.
__global__ void k_aggregate(const float* __restrict__ h, const float* __restrict__ alpha,
                            const int* __restrict__ csr, const int* __restrict__ off,
                            const int* __restrict__ src, const float* __restrict__ bias,
                            float* __restrict__ xout, int M) {
    int wave = (blockIdx.x * blockDim.x + threadIdx.x) >> 5;
    if (wave >= cN) return;                    // wave-uniform
    int lane = threadIdx.x & 31;
    int jn = M >> 5;                           // 8 (M=256) or 4 (M=128)
    float acc[8];
#pragma unroll
    for (int j = 0; j < 8; ++j) acc[j] = 0.f;
    int s0 = off[wave], s1 = off[wave + 1];
    for (int i = s0; i < s1; ++i) {
        int e = csr[i];
        float a = alpha[e];
        const float* hp = h + (size_t)src[e] * M;
        for (int j = 0; j < jn; ++j) acc[j] += a * hp[lane + 32 * j];
    }
    float* op = xout + (size_t)wave * M;
    for (int j = 0; j < jn; ++j) {
        float v = acc[j] + bias[lane + 32 * j];
        op[lane + 32 * j] = (v > 0.f) ? v : 0.f;
    }
}

// ------------------------------------------------------ fused edge-MLP head
// One wave per 16-edge tile: hcat gather + PReLU + [16x128]@[128x128] WMMA
// + PReLU + [*]@[128x2] + softmax + message.
__global__ void k_head_edges(const float* __restrict__ ys, const float* __restrict__ yd,
                             const int* __restrict__ src, const int* __restrict__ dst,
                             const float* __restrict__ a1, const float* __restrict__ W1,
                             const float* __restrict__ b1, const float* __restrict__ a2,
                             const float* __restrict__ W2, const float* __restrict__ b2,
                             const float* __restrict__ raw, float* __restrict__ pred_conn,
                             float* __restrict__ prob_conn, float* __restrict__ msg) {
    int wave = (blockIdx.x * blockDim.x + threadIdx.x) >> 5;
    if (wave >= cE / 16) return;               // wave-uniform
    int lane = threadIdx.x & 31;
    int half = lane >> 4;
    int l16  = lane & 15;
    int row0 = wave * 16;
    int er   = row0 + l16;                     // this lane's A-row (edge)
    const float* ysp = ys + (size_t)src[er] * cNH;
    const float* ydp = yd + (size_t)dst[er] * cNH;

    v8f acc[8];
#pragma unroll
    for (int t = 0; t < 8; ++t) acc[t] = (v8f){0.f,0.f,0.f,0.f,0.f,0.f,0.f,0.f};

    for (int k0 = 0; k0 < cNH; k0 += 4) {
        v2f a;
#pragma unroll
        for (int j = 0; j < 2; ++j) {
            int c = k0 + half * 2 + j;
            float t = ysp[c] + ydp[c];          // hcat
            a[j] = (t >= 0.f) ? t : a1[c] * t;  // PReLU(hcat, a1)
        }
#pragma unroll
        for (int t = 0; t < 8; ++t) {
            int cc = t * 16 + l16;
            v2f b;
            b[0] = W1[(size_t)(k0 + half * 2 + 0) * cNH + cc];
            b[1] = W1[(size_t)(k0 + half * 2 + 1) * cNH + cc];
            acc[t] = __builtin_amdgcn_wmma_f32_16x16x4_f32(
                false, a, false, b, (short)0, acc[t], false, false);
        }
    }

    // u = PReLU(acc + b1, a2); fold into 2 logits via W2
    float p0[8], p1[8];
#pragma unroll
    for (int r = 0; r < 8; ++r) { p0[r] = 0.f; p1[r] = 0.f; }
#pragma unroll
    for (int t = 0; t < 8; ++t) {
        int c = t * 16 + l16;
        float bb = b1[c], aa = a2[c];
        float w20 = W2[c * 2 + 0], w21 = W2[c * 2 + 1];
#pragma unroll
        for (int r = 0; r < 8; ++r) {
            float v = acc[t][r] + bb;
            v = (v >= 0.f) ? v : aa * v;
            p0[r] += v * w20;
            p1[r] += v * w21;
        }
    }
    // butterfly-reduce over the 16 lanes of each half (rows differ by half)
#pragma unroll
    for (int m = 1; m < 16; m <<= 1) {
#pragma unroll
        for (int r = 0; r < 8; ++r) {
            p0[r] += __shfl_xor(p0[r], m, 32);
            p1[r] += __shfl_xor(p1[r], m, 32);
        }
    }
    float bb0 = b2[0], bb1 = b2[1];
    if (l16 < 8) {
        int r = l16;
        int edge = row0 + half * 8 + r;
        float q0 = p0[r] + bb0, q1 = p1[r] + bb1;
        pred_conn[edge * 2 + 0] = q0;
        pred_conn[edge * 2 + 1] = q1;
        float mx = fmaxf(q0, q1);
        float e0 = __expf(q0 - mx), e1 = __expf(q1 - mx);
        float s = e0 + e1;
        float pr0 = e0 / s, pr1 = e1 / s;
        prob_conn[edge * 2 + 0] = pr0;
        prob_conn[edge * 2 + 1] = pr1;
        msg[edge] = raw[edge] * (pr1 - pr0);
    }
}

__global__ void k_pred_den(const float* __restrict__ msg, const int* __restrict__ csr,
                           const int* __restrict__ off, float* __restrict__ pden) {
    int n = blockIdx.x * blockDim.x + threadIdx.x;
    if (n >= cN) return;
    int s0 = off[n], s1 = off[n + 1];
    float s = 0.f;
    for (int i = s0; i < s1; ++i) s += msg[csr[i]];
    float deg = (float)(s1 - s0);
    pden[n] = s / fmaxf(deg, 1.0f);
}

// ---------------------------------------------------------------------------
extern "C" void kernel_launch(void* const* d_in, const int* in_sizes, int n_in,
                              void* d_out, int out_size, void* d_ws, size_t ws_size,
                              hipStream_t stream) {
    const float* feat  = (const float*)d_in[0];
    const float* cfeat = (const float*)d_in[1];
    const int*   srcp  = (const int*)d_in[2];
    const int*   dstp  = (const int*)d_in[3];
    const float* raw   = (const float*)d_in[4];
    // gat_params: 4 x (W, b, al, ar) at indices 5..20
    // head_params at 21..30: Ws bs Wd bd a1 W1 b1 a2 W2 b2
    const float* Ws = (const float*)d_in[21];
    const float* bs = (const float*)d_in[22];
    const float* Wd = (const float*)d_in[23];
    const float* bd = (const float*)d_in[24];
    const float* a1 = (const float*)d_in[25];
    const float* W1 = (const float*)d_in[26];
    const float* b1 = (const float*)d_in[27];
    const float* a2 = (const float*)d_in[28];
    const float* W2 = (const float*)d_in[29];
    const float* b2 = (const float*)d_in[30];

    // ---- workspace carve-up (256B aligned) ----
    char* base = (char*)d_ws;
    size_t o = 0;
    auto carve = [&](size_t bytes) -> char* {
        char* p = base + o;
        o = (o + bytes + 255) & ~(size_t)255;
        return p;
    };
    float* bufA  = (float*)carve((size_t)cN * 512 * 4);   // x0 / ping-pong
    float* bufB  = (float*)carve((size_t)cN * 256 * 4);   // ping-pong
    float* hbuf  = (float*)carve((size_t)cN * 256 * 4);   // h per layer
    float* ysb   = (float*)carve((size_t)cN * cNH * 4);
    float* ydb   = (float*)carve((size_t)cN * cNH * 4);
    float* elb   = (float*)carve((size_t)cN * 4);
    float* erb   = (float*)carve((size_t)cN * 4);
    float* mb    = (float*)carve((size_t)cN * 4);
    float* zb    = (float*)carve((size_t)cN * 4);
    float* eedge = (float*)carve((size_t)cE * 4);
    float* alph  = (float*)carve((size_t)cE * 4);
    float* msgb  = (float*)carve((size_t)cE * 4);
    int*   deg   = (int*)carve((size_t)cN * 4);
    int*   cur   = (int*)carve((size_t)cN * 4);
    int*   off   = (int*)carve((size_t)(cN + 1) * 4);
    int*   csr   = (int*)carve((size_t)cE * 4);

    float* pred_conn = (float*)d_out;                   // [E,2]
    float* prob_conn = (float*)d_out + 2 * (size_t)cE;  // [E,2]
    float* pden      = (float*)d_out + 4 * (size_t)cE;  // [N]

    // ---- graph structure (CSR by dst), rebuilt deterministically each call
    k_zero_i32<<<(cN + 255) / 256, 256, 0, stream>>>(deg, cN);
    k_zero_i32<<<(cN + 255) / 256, 256, 0, stream>>>(cur, cN);
    k_deg<<<cE / 256, 256, 0, stream>>>(dstp, deg);
    k_scan<<<1, 1024, 0, stream>>>(deg, off, cN);
    k_place<<<cE / 256, 256, 0, stream>>>(dstp, off, cur, csr);

    // ---- input concat ----
    k_concat<<<(cN * 512 + 255) / 256, 256, 0, stream>>>(feat, cfeat, bufA);

    // ---- 4 GAT layers ----
    const int Kd[4] = {512, 256, 256, 128};
    const int Md[4] = {256, 256, 128, 128};
    float* xin = bufA;
    for (int l = 0; l < 4; ++l) {
        const float* W  = (const float*)d_in[5 + 4 * l];
        const float* b  = (const float*)d_in[6 + 4 * l];
        const float* al = (const float*)d_in[7 + 4 * l];
        const float* ar = (const float*)d_in[8 + 4 * l];
        int K = Kd[l], M = Md[l];
        float* xout = (xin == bufA) ? bufB : bufA;

        int tiles = (cN / 16) * (M / 16);
        k_gemm<<<(tiles + 3) / 4, 128, 0, stream>>>(xin, W, nullptr, hbuf, K, M);
        k_scores<<<(cN + 255) / 256, 256, 0, stream>>>(hbuf, al, ar, elb, erb, M);
        k_logits<<<cE / 256, 256, 0, stream>>>(elb, erb, srcp, dstp, eedge);
        k_segmax_z<<<(cN + 255) / 256, 256, 0, stream>>>(eedge, csr, off, mb, zb);
        k_alpha<<<cE / 256, 256, 0, stream>>>(eedge, mb, zb, dstp, alph);
        k_aggregate<<<(cN * 32 + 255) / 256, 256, 0, stream>>>(hbuf, alph, csr, off,
                                                               srcp, b, xout, M);
        xin = xout;
    }
    // xin == bufA holds conv_features [N, 128]

    // ---- head node-level MLPs ----
    {
        int tiles = (cN / 16) * (cNH / 16);
        k_gemm<<<(tiles + 3) / 4, 128, 0, stream>>>(xin, Ws, bs, ysb, cNH, cNH);
        k_gemm<<<(tiles + 3) / 4, 128, 0, stream>>>(xin, Wd, bd, ydb, cNH, cNH);
    }

    // ---- fused per-edge head ----
    k_head_edges<<<(cE / 16 + 3) / 4, 128, 0, stream>>>(
        ysb, ydb, srcp, dstp, a1, W1, b1, a2, W2, b2, raw,
        pred_conn, prob_conn, msgb);

    // ---- pred_den ----
    k_pred_den<<<(cN + 255) / 256, 256, 0, stream>>>(msgb, csr, off, pden);
}